// ExpABPropagation_82420422410744
// MI455X (gfx1250) — compile-verified
//
#include <hip/hip_runtime.h>
#include <stdint.h>

#define WAVE 32
#define WPB  8   // waves per block in spmm

#if defined(__has_builtin)
#if __has_builtin(__builtin_amdgcn_global_load_async_to_lds_b64) && \
    __has_builtin(__builtin_amdgcn_s_wait_asynccnt)
#define HAVE_ASYNC_LDS 1
#endif
#endif
#ifndef HAVE_ASYNC_LDS
#define HAVE_ASYNC_LDS 0
#endif

#if HAVE_ASYNC_LDS
typedef int v2i __attribute__((ext_vector_type(2)));
typedef __attribute__((address_space(1))) v2i* g_v2i_ptr;
typedef __attribute__((address_space(3))) v2i* l_v2i_ptr;

static __device__ __forceinline__ g_v2i_ptr to_global_v2i(const void* p) {
    return (g_v2i_ptr)(uintptr_t)p;               // generic global addr == AS1 addr
}
static __device__ __forceinline__ l_v2i_ptr to_lds_v2i(const void* p) {
    return (l_v2i_ptr)(uint32_t)(uintptr_t)p;     // low 32 bits of generic LDS addr == LDS offset
}
#endif

// ---------------- CSR build ----------------

__global__ void count_deg_kernel(const long long* __restrict__ rows,
                                 int* __restrict__ counts, int E) {
    int e = blockIdx.x * blockDim.x + threadIdx.x;
    if (e < E) atomicAdd(&counts[(int)rows[e]], 1);
}

__global__ void dis_kernel(const int* __restrict__ counts,
                           float* __restrict__ dis, int N) {
    int n = blockIdx.x * blockDim.x + threadIdx.x;
    if (n < N) {
        int d = counts[n];
        // deg>0 ? rsqrt(max(deg,1)) : 0  ==  deg>0 ? 1/sqrt(deg) : 0
        dis[n] = (d > 0) ? (1.0f / sqrtf((float)d)) : 0.0f;
    }
}

// Single-workgroup exclusive scan over N counts -> offsets[0..N]
__global__ void scan_kernel(const int* __restrict__ counts,
                            int* __restrict__ offsets, int N) {
    __shared__ int buf[1024];
    __shared__ int carry;
    int t = threadIdx.x;
    if (t == 0) carry = 0;
    __syncthreads();
    for (int base = 0; base < N; base += 1024) {
        int i = base + t;
        int v = (i < N) ? counts[i] : 0;
        buf[t] = v;
        __syncthreads();
        for (int off = 1; off < 1024; off <<= 1) {
            int add = (t >= off) ? buf[t - off] : 0;
            __syncthreads();
            buf[t] += add;
            __syncthreads();
        }
        if (i < N) offsets[i] = carry + (buf[t] - v);   // exclusive
        __syncthreads();
        if (t == 0) carry += buf[1023];
        __syncthreads();
    }
    if (t == 0) offsets[N] = carry;
}

// Pack (col, -eta*dis[row]*dis[col]*mask[col]) into CSR slots.
__global__ void scatter_kernel(const long long* __restrict__ rows,
                               const long long* __restrict__ cols,
                               const float* __restrict__ dis,
                               const int* __restrict__ mask,
                               const int* __restrict__ offsets,
                               int* __restrict__ cursor,
                               uint2* __restrict__ edges, int E) {
    int e = blockIdx.x * blockDim.x + threadIdx.x;
    if (e < E) {
        int r = (int)rows[e];
        int c = (int)cols[e];
        float w = -1.0f * dis[r] * dis[c] * (float)mask[c];   // ETA = 1.0
        int pos = offsets[r] + atomicAdd(&cursor[r], 1);
        edges[pos] = make_uint2((unsigned)c, __float_as_uint(w));
    }
}

// term = residuals ; acc(d_out) = residuals
__global__ void init_kernel(const float4* __restrict__ src,
                            float4* __restrict__ term,
                            float4* __restrict__ acc, int n4) {
    int i = blockIdx.x * blockDim.x + threadIdx.x;
    if (i < n4) {
        float4 v = src[i];
        term[i] = v;
        acc[i]  = v;
    }
}

// ---------------- Hot loop: one wave32 per destination node, C==64 ----------------
// Edge records (col, weight) are streamed into per-wave LDS slots with
// GLOBAL_LOAD_ASYNC_TO_LDS_B64, double-buffered on ASYNCcnt; compute overlaps
// with the async engine. Each lane owns channels [2*lane, 2*lane+1].
__global__ void spmm_kernel(const float* __restrict__ x,
                            float* __restrict__ term_out,
                            float* __restrict__ acc,
                            const uint2* __restrict__ edges,
                            const int* __restrict__ offsets,
                            float inv_k, int N) {
    int lane = threadIdx.x & (WAVE - 1);
    int w    = threadIdx.x >> 5;
    int n    = blockIdx.x * WPB + w;
    if (n >= N) return;

    int start = offsets[n];
    int end   = offsets[n + 1];
    float sx = 0.0f, sy = 0.0f;

#if HAVE_ASYNC_LDS
    __shared__ uint2 sbuf[WPB][2][WAVE];
    int buf = 0;
    if (start < end) {
        int j = start + lane;
        if (j < end)
            __builtin_amdgcn_global_load_async_to_lds_b64(
                to_global_v2i(edges + j), to_lds_v2i(&sbuf[w][0][lane]), 0, 0);
    }
    for (int base = start; base < end; base += WAVE) {
        int nextb = base + WAVE;
        if (nextb < end) {
            int j = nextb + lane;
            if (j < end)
                __builtin_amdgcn_global_load_async_to_lds_b64(
                    to_global_v2i(edges + j), to_lds_v2i(&sbuf[w][buf ^ 1][lane]), 0, 0);
            __builtin_amdgcn_s_wait_asynccnt(1);   // batch `base` has landed (in-order)
        } else {
            __builtin_amdgcn_s_wait_asynccnt(0);
        }
        __asm__ volatile("" ::: "memory");          // async wrote LDS behind compiler's back
        int cnt = min(WAVE, end - base);
        for (int i = 0; i < cnt; ++i) {
            uint2 r = sbuf[w][buf][i];              // uniform addr -> ds_load broadcast
            int   cj = __builtin_amdgcn_readfirstlane((int)r.x);
            float wj = __uint_as_float(__builtin_amdgcn_readfirstlane((int)r.y));
            const float2 xv = *(const float2*)(x + (size_t)(unsigned)cj * 64 + lane * 2);
            sx = fmaf(wj, xv.x, sx);
            sy = fmaf(wj, xv.y, sy);
        }
        buf ^= 1;
    }
#else
    for (int base = start; base < end; base += WAVE) {
        int j = base + lane;
        uint2 rec = make_uint2(0u, 0u);
        if (j < end) rec = edges[j];
        if (base + WAVE < end)                      // gfx1250 global_prefetch_b8
            __builtin_prefetch(&edges[base + WAVE + lane], 0, 0);
        int cnt = min(WAVE, end - base);
        #pragma unroll 4
        for (int i = 0; i < cnt; ++i) {
            unsigned cj = (unsigned)__builtin_amdgcn_readlane((int)rec.x, i);
            float    wj = __uint_as_float(__builtin_amdgcn_readlane((int)rec.y, i));
            const float2 xv = *(const float2*)(x + (size_t)cj * 64 + lane * 2);
            sx = fmaf(wj, xv.x, sx);
            sy = fmaf(wj, xv.y, sy);
        }
    }
#endif

    sx *= inv_k; sy *= inv_k;

    float2* to = (float2*)(term_out + (size_t)n * 64) + lane;
    *to = make_float2(sx, sy);
    float2* ao = (float2*)(acc + (size_t)n * 64) + lane;
    float2 av = *ao;
    av.x += sx; av.y += sy;
    *ao = av;
}

// ---------------- Host ----------------

static inline char* alignup(char* p, size_t a) {
    return (char*)(((uintptr_t)p + (a - 1)) & ~(uintptr_t)(a - 1));
}

extern "C" void kernel_launch(void* const* d_in, const int* in_sizes, int n_in,
                              void* d_out, int out_size, void* d_ws, size_t ws_size,
                              hipStream_t stream) {
    const float*     residuals  = (const float*)d_in[0];
    const long long* edge_index = (const long long*)d_in[1];   // int64
    const int*       train_mask = (const int*)d_in[2];

    const int N = in_sizes[2];
    const int E = in_sizes[1] / 2;
    const int C = in_sizes[0] / N;   // 64

    const long long* rows = edge_index;
    const long long* cols = edge_index + E;

    // workspace layout
    char* p = (char*)d_ws;
    int* counts = (int*)p;  p += (size_t)N * 4;
    int* cursor = (int*)p;  p += (size_t)N * 4;     // contiguous with counts
    p = alignup(p, 256);
    int* offsets = (int*)p; p += (size_t)(N + 1) * 4;
    p = alignup(p, 256);
    float* dis = (float*)p; p += (size_t)N * 4;
    p = alignup(p, 256);
    uint2* edges = (uint2*)p; p += (size_t)E * 8;
    p = alignup(p, 256);
    float* termA = (float*)p; p += (size_t)N * C * 4;
    p = alignup(p, 256);
    float* termB = (float*)p;

    float* acc = (float*)d_out;

    (void)hipMemsetAsync(counts, 0, (size_t)N * 8, stream);  // counts + cursor

    count_deg_kernel<<<(E + 255) / 256, 256, 0, stream>>>(rows, counts, E);
    dis_kernel<<<(N + 255) / 256, 256, 0, stream>>>(counts, dis, N);
    scan_kernel<<<1, 1024, 0, stream>>>(counts, offsets, N);
    scatter_kernel<<<(E + 255) / 256, 256, 0, stream>>>(rows, cols, dis, train_mask,
                                                        offsets, cursor, edges, E);
    int n4 = (N * C) / 4;
    init_kernel<<<(n4 + 255) / 256, 256, 0, stream>>>((const float4*)residuals,
                                                      (float4*)termA, (float4*)acc, n4);

    const float* tin = termA;
    float* tout = termB;
    for (int k = 1; k <= 40; ++k) {
        float inv_k = 1.0f / (float)k;
        spmm_kernel<<<(N + WPB - 1) / WPB, 256, 0, stream>>>(tin, tout, acc, edges,
                                                             offsets, inv_k, N);
        const float* t = tout; tout = (float*)tin; tin = t;
    }
}